// GNNTracker_51384988729466
// MI455X (gfx1250) — compile-verified
//
#include <hip/hip_runtime.h>
#include <cstdint>
#include <cstddef>

typedef __attribute__((ext_vector_type(2))) float v2f;
typedef __attribute__((ext_vector_type(8))) float v8f;

static __device__ __forceinline__ unsigned short f2bf(float f) {
  unsigned u = __float_as_uint(f);
  unsigned r = (u >> 16) & 1u;
  return (unsigned short)((u + 0x7FFFu + r) >> 16);
}
static __device__ __forceinline__ unsigned encOrd(float f) {
  unsigned u = __float_as_uint(f);
  return (u >> 31) ? ~u : (u | 0x80000000u);
}
static __device__ __forceinline__ float decOrd(unsigned u) {
  return (u >> 31) ? __uint_as_float(u & 0x7FFFFFFFu) : __uint_as_float(~u);
}
// Native f32 global atomic add (avoid CAS fallback).
static __device__ __forceinline__ void atomAddF32(float* p, float v) {
  asm volatile("global_atomic_add_f32 %0, %1, off"
               :: "v"((uint64_t)(uintptr_t)p), "v"(v) : "memory");
}

// ---------------------------------------------------------------------------
// WMMA f32 16x16x4 GEMM, compile-time K and NT=Nout/16:
//   Y[M, NT*16] = op(X)[M, K] @ W[K, NT*16] + bias
// One wave computes a 16-row strip across all NT column tiles. Full K unroll
// -> immediate-offset loads, clause-batched, overlapping the WMMA chain.
// flags: bit0 = batch-norm(stats)+ReLU applied to X on load
//        bit1 = bf16 output, bit2 = accumulate into Y (f32)
// ---------------------------------------------------------------------------
template <int K, int NT>
__global__ __launch_bounds__(128) void k_gemm_wmma(
    const float* __restrict__ X, const float* __restrict__ W,
    const float* __restrict__ bias, void* __restrict__ Yo,
    int Mtiles,
    const float* __restrict__ stats, const float* __restrict__ gam,
    const float* __restrict__ bet, float invRows, int flags)
{
  constexpr int Nout = NT * 16;
  const int wave = threadIdx.x >> 5;
  const int lane = threadIdx.x & 31;
  const int mt = blockIdx.x * 4 + wave;
  if (mt >= Mtiles) return;                 // whole-wave exit, EXEC stays full
  const int l15 = lane & 15;
  const int half = lane >> 4;               // 0: K 0,1   1: K 2,3
  const int row = mt * 16 + l15;

  const float* Xr = X + (size_t)row * K + half * 2;  // this lane's A base
  const float* Wb = W + (size_t)(half * 2) * Nout + l15;

  v8f zero = {0.f,0.f,0.f,0.f,0.f,0.f,0.f,0.f};
  v8f acc[NT];
#pragma unroll
  for (int j = 0; j < NT; ++j) acc[j] = zero;

  const int doBN = flags & 1;
#pragma unroll
  for (int k = 0; k < K; k += 4) {
    const int ka = k + half * 2;
    float2 ap = *(const float2*)(Xr + k);            // one b64 load
    float a0 = ap.x, a1 = ap.y;
    if (doBN) {
      float mu0 = stats[ka] * invRows;
      float mu1 = stats[ka + 1] * invRows;
      float vr0 = stats[K + ka] * invRows - mu0 * mu0;
      float vr1 = stats[K + ka + 1] * invRows - mu1 * mu1;
      a0 = fmaxf((a0 - mu0) * rsqrtf(vr0 + 1e-5f) * gam[ka] + bet[ka], 0.f);
      a1 = fmaxf((a1 - mu1) * rsqrtf(vr1 + 1e-5f) * gam[ka + 1] + bet[ka + 1], 0.f);
    }
    v2f av = {a0, a1};
    const float* Wr = Wb + (size_t)k * Nout;
#pragma unroll
    for (int j = 0; j < NT; ++j) {
      v2f bv = {Wr[j * 16], Wr[Nout + j * 16]};
      acc[j] = __builtin_amdgcn_wmma_f32_16x16x4_f32(
          false, av, false, bv, (short)0, acc[j], false, false);
    }
  }

  const int rbase = mt * 16 + half * 8;
#pragma unroll
  for (int j = 0; j < NT; ++j) {
    const int nc = j * 16 + l15;
    const float bc = bias[nc];
    if (flags & 2) {
      unsigned short* Y = (unsigned short*)Yo;
#pragma unroll
      for (int r = 0; r < 8; ++r)
        Y[(size_t)(rbase + r) * Nout + nc] = f2bf(acc[j][r] + bc);
    } else if (flags & 4) {
      float* Y = (float*)Yo;
#pragma unroll
      for (int r = 0; r < 8; ++r)
        Y[(size_t)(rbase + r) * Nout + nc] += acc[j][r] + bc;
    } else {
      float* Y = (float*)Yo;
#pragma unroll
      for (int r = 0; r < 8; ++r)
        Y[(size_t)(rbase + r) * Nout + nc] = acc[j][r] + bc;
    }
  }
}

// ---------------------------------------------------------------------------
// Column sums + sum-of-squares (for batch-norm batch statistics).
// ---------------------------------------------------------------------------
__global__ __launch_bounds__(256) void k_colstats(
    const float* __restrict__ X, int M, int Cc, float* __restrict__ stats)
{
  __shared__ float ps[256], pq[256];
  const int c = threadIdx.x % Cc;
  const int rl = threadIdx.x / Cc;
  const int rstep = 256 / Cc;
  float s = 0.f, q = 0.f;
  for (long r = (long)blockIdx.x * rstep + rl; r < M; r += (long)gridDim.x * rstep) {
    float v = X[r * Cc + c];
    s += v; q += v * v;
  }
  ps[threadIdx.x] = s; pq[threadIdx.x] = q;
  __syncthreads();
  if (threadIdx.x < Cc) {
    float S = 0.f, Q = 0.f;
    for (int r = 0; r < rstep; ++r) { S += ps[c + r * Cc]; Q += pq[c + r * Cc]; }
    atomAddF32(&stats[c], S);
    atomAddF32(&stats[Cc + c], Q);
  }
}

// ---------------------------------------------------------------------------
// Row LayerNorm + ReLU (blockDim == Cc, one row per block). Y may alias X.
// ---------------------------------------------------------------------------
__global__ void k_ln_relu(const float* __restrict__ X, float* __restrict__ Y,
                          const float* __restrict__ g, const float* __restrict__ b,
                          int Cc)
{
  __shared__ float s1[128], s2[128];
  const int row = blockIdx.x, c = threadIdx.x;
  float v = X[(size_t)row * Cc + c];
  s1[c] = v; s2[c] = v * v;
  __syncthreads();
  for (int o = Cc >> 1; o > 0; o >>= 1) {
    if (c < o) { s1[c] += s1[c + o]; s2[c] += s2[c + o]; }
    __syncthreads();
  }
  float mu = s1[0] / (float)Cc;
  float var = s2[0] / (float)Cc - mu * mu;
  float y = (v - mu) * rsqrtf(var + 1e-5f) * g[c] + b[c];
  Y[(size_t)row * Cc + c] = fmaxf(y, 0.f);
}

// ---------------------------------------------------------------------------
// Edge pass 1: tm = t * (relu(h[src]+e) + eps); ordered-u32 atomic max per
// (dst,feature). One wave per edge, lane covers 2 features.
// ---------------------------------------------------------------------------
__global__ __launch_bounds__(256) void k_edge_pass1(
    const float* __restrict__ h, const unsigned short* __restrict__ e,
    const int* __restrict__ src, const int* __restrict__ dst,
    const float* __restrict__ ct, int layer, unsigned* __restrict__ mx, int nE)
{
  const int wid = (blockIdx.x * blockDim.x + threadIdx.x) >> 5;
  if (wid >= nE) return;
  const int lane = threadIdx.x & 31;
  const float t = ct[layer];
  const int s = __builtin_amdgcn_readfirstlane(src[wid]);
  const int d = __builtin_amdgcn_readfirstlane(dst[wid]);
  const int c2 = lane * 2;
  float2 hx = *(const float2*)(h + (size_t)s * 64 + c2);
  unsigned ee = *(const unsigned*)(e + (size_t)wid * 64 + c2);
  float e0 = __uint_as_float(ee << 16);
  float e1 = __uint_as_float(ee & 0xFFFF0000u);
  float m0 = fmaxf(hx.x + e0, 0.f) + 1e-7f;
  float m1 = fmaxf(hx.y + e1, 0.f) + 1e-7f;
  unsigned* p = mx + (size_t)d * 64 + c2;
  atomicMax(p, encOrd(t * m0));
  atomicMax(p + 1, encOrd(t * m1));
}

// ---------------------------------------------------------------------------
// Edge pass 2: w = exp(tm - mx[dst]); num[dst] += msg*w; den[dst] += w.
// ---------------------------------------------------------------------------
__global__ __launch_bounds__(256) void k_edge_pass2(
    const float* __restrict__ h, const unsigned short* __restrict__ e,
    const int* __restrict__ src, const int* __restrict__ dst,
    const float* __restrict__ ct, int layer, const unsigned* __restrict__ mx,
    float* __restrict__ num, float* __restrict__ den, int nE)
{
  const int wid = (blockIdx.x * blockDim.x + threadIdx.x) >> 5;
  if (wid >= nE) return;
  const int lane = threadIdx.x & 31;
  const float t = ct[layer];
  const int s = __builtin_amdgcn_readfirstlane(src[wid]);
  const int d = __builtin_amdgcn_readfirstlane(dst[wid]);
  const int c2 = lane * 2;
  float2 hx = *(const float2*)(h + (size_t)s * 64 + c2);
  unsigned ee = *(const unsigned*)(e + (size_t)wid * 64 + c2);
  float e0 = __uint_as_float(ee << 16);
  float e1 = __uint_as_float(ee & 0xFFFF0000u);
  float m0 = fmaxf(hx.x + e0, 0.f) + 1e-7f;
  float m1 = fmaxf(hx.y + e1, 0.f) + 1e-7f;
  size_t off = (size_t)d * 64 + c2;
  float w0 = __expf(t * m0 - decOrd(mx[off]));
  float w1 = __expf(t * m1 - decOrd(mx[off + 1]));
  atomAddF32(num + off, m0 * w0);
  atomAddF32(num + off + 1, m1 * w1);
  atomAddF32(den + off, w0);
  atomAddF32(den + off + 1, w1);
}

// out[i] = num[i] / (den[i] + 1e-16) + xin[i]   (out may alias num)
__global__ void k_aggr(const float* __restrict__ num, const float* __restrict__ den,
                       const float* __restrict__ xin, float* __restrict__ outp, int n)
{
  for (int i = blockIdx.x * blockDim.x + threadIdx.x; i < n;
       i += gridDim.x * blockDim.x)
    outp[i] = num[i] / (den[i] + 1e-16f) + xin[i];
}

// Final: out[i,:] = relu(bn(t16[i,:])) @ oW3[16,2] + ob3
__global__ void k_final(const float* __restrict__ t16, const float* __restrict__ stats,
                        const float* __restrict__ g, const float* __restrict__ b,
                        const float* __restrict__ W3, const float* __restrict__ b3,
                        float* __restrict__ out, int n, float invRows)
{
  int i = blockIdx.x * blockDim.x + threadIdx.x;
  if (i >= n) return;
  float o0 = b3[0], o1 = b3[1];
#pragma unroll
  for (int c = 0; c < 16; ++c) {
    float mu = stats[c] * invRows;
    float var = stats[16 + c] * invRows - mu * mu;
    float v = (t16[(size_t)i * 16 + c] - mu) * rsqrtf(var + 1e-5f) * g[c] + b[c];
    v = fmaxf(v, 0.f);
    o0 += v * W3[c * 2 + 0];
    o1 += v * W3[c * 2 + 1];
  }
  out[2 * i] = o0;
  out[2 * i + 1] = o1;
}

// ---------------------------------------------------------------------------
static void launch_gemm(hipStream_t st, const float* X, const float* W, const float* b,
                        void* Y, int M, int K, int Nout,
                        const float* stats, const float* g, const float* be,
                        float invRows, int flags)
{
  int Mtiles = M / 16;
  int blocks = (Mtiles + 3) / 4;
#define GEMM_CASE(KK, NTT)                                                   \
  if (K == KK && Nout == NTT * 16) {                                         \
    k_gemm_wmma<KK, NTT><<<blocks, 128, 0, st>>>(X, W, b, Y, Mtiles,         \
                                                 stats, g, be, invRows, flags); \
    return;                                                                  \
  }
  GEMM_CASE(16, 4)
  GEMM_CASE(64, 4)
  GEMM_CASE(8, 4)
  GEMM_CASE(64, 8)
  GEMM_CASE(128, 4)
  GEMM_CASE(64, 2)
  GEMM_CASE(32, 1)
#undef GEMM_CASE
}

extern "C" void kernel_launch(void* const* d_in, const int* in_sizes, int n_in,
                              void* d_out, int out_size, void* d_ws, size_t ws_size,
                              hipStream_t stream)
{
  (void)n_in; (void)out_size; (void)ws_size;
  const float* x   = (const float*)d_in[0];
  const float* ea  = (const float*)d_in[1];
  const int*   ei  = (const int*)d_in[2];
  const float* nW1 = (const float*)d_in[3];  const float* nb1 = (const float*)d_in[4];
  const float* ng1 = (const float*)d_in[5];  const float* nB1 = (const float*)d_in[6];
  const float* nW2 = (const float*)d_in[7];  const float* nb2 = (const float*)d_in[8];
  const float* eW1 = (const float*)d_in[9];  const float* eb1 = (const float*)d_in[10];
  const float* eg1 = (const float*)d_in[11]; const float* eB1 = (const float*)d_in[12];
  const float* eW2 = (const float*)d_in[13]; const float* eb2 = (const float*)d_in[14];
  const float* lng = (const float*)d_in[15]; const float* lnb = (const float*)d_in[16];
  const float* cW1 = (const float*)d_in[17]; const float* cb1 = (const float*)d_in[18];
  const float* cg  = (const float*)d_in[19]; const float* cB  = (const float*)d_in[20];
  const float* cW2 = (const float*)d_in[21]; const float* cb2 = (const float*)d_in[22];
  const float* ct  = (const float*)d_in[23];
  const float* oW1 = (const float*)d_in[24]; const float* ob1 = (const float*)d_in[25];
  const float* og1 = (const float*)d_in[26]; const float* oB1 = (const float*)d_in[27];
  const float* oW2 = (const float*)d_in[28]; const float* ob2 = (const float*)d_in[29];
  const float* og2 = (const float*)d_in[30]; const float* oB2 = (const float*)d_in[31];
  const float* oW3 = (const float*)d_in[32]; const float* ob3 = (const float*)d_in[33];

  const int N = in_sizes[0] / 16;   // 50000
  const int E = in_sizes[1] / 8;    // 800000
  const int* srcI = ei;
  const int* dstI = ei + E;

  // ---- workspace layout (~320 MB) ----
  uint8_t* ws = (uint8_t*)d_ws;
  const size_t eB_bytes = (size_t)E * 64 * 2;        // 102.4 MB bf16 edge feats
  const size_t h_bytes  = (size_t)N * 64 * 4;        // 12.8 MB
  unsigned short* eB = (unsigned short*)ws;
  float* h = (float*)(ws + eB_bytes);
  uint8_t* big = ws + eB_bytes + h_bytes;            // E*64*4 = 204.8 MB region
  float*    tBig = (float*)big;                      // encoder pre-BN temp (whole region)
  float*    num  = (float*)big;                      // later: [N,64]
  float*    den  = (float*)(big + h_bytes);
  unsigned* mx   = (unsigned*)(big + 2 * h_bytes);
  float*    t128 = (float*)(big + 3 * h_bytes);      // [N,128]
  float*    hb   = (float*)(big + 3 * h_bytes + (size_t)N * 128 * 4); // [N,64]
  float*    t32  = t128;                             // [N,32]
  float*    t16  = (float*)((uint8_t*)t128 + (size_t)N * 32 * 4);     // [N,16]
  float*    stats0 = (float*)(big + (size_t)E * 64 * 4);
  float*    stats1 = stats0 + 256;

  // ---- node encoder: h = relu(bn(x@nW1+nb1)) @ nW2 + nb2 ----
  launch_gemm(stream, x, nW1, nb1, tBig, N, 16, 64, nullptr, nullptr, nullptr, 0.f, 0);
  hipMemsetAsync(stats0, 0, 2 * 64 * 4, stream);
  k_colstats<<<1024, 256, 0, stream>>>(tBig, N, 64, stats0);
  launch_gemm(stream, tBig, nW2, nb2, h, N, 64, 64, stats0, ng1, nB1, 1.f / (float)N, 1);

  // ---- edge encoder: e = relu(bn(ea@eW1+eb1)) @ eW2 + eb2  (bf16 out) ----
  launch_gemm(stream, ea, eW1, eb1, tBig, E, 8, 64, nullptr, nullptr, nullptr, 0.f, 0);
  hipMemsetAsync(stats1, 0, 2 * 64 * 4, stream);
  k_colstats<<<2048, 256, 0, stream>>>(tBig, E, 64, stats1);
  launch_gemm(stream, tBig, eW2, eb2, eB, E, 64, 64, stats1, eg1, eB1, 1.f / (float)E, 1 | 2);

  const int edgeBlocks = (E * 32 + 255) / 256;       // one wave per edge

  for (int l = 0; l < 4; ++l) {
    const float* sx;
    if (l == 0) {
      sx = h;                                        // layer0: conv applied to h directly
    } else {
      k_ln_relu<<<N, 64, 0, stream>>>(h, hb, lng + 64 * l, lnb + 64 * l, 64);
      sx = hb;                                       // res+: r = relu(LN(h))
    }
    hipMemsetAsync(num, 0, h_bytes, stream);
    hipMemsetAsync(den, 0, h_bytes, stream);
    hipMemsetAsync(mx, 0, h_bytes, stream);          // 0 < encOrd(any real tm)
    k_edge_pass1<<<edgeBlocks, 256, 0, stream>>>(sx, eB, srcI, dstI, ct, l, mx, E);
    k_edge_pass2<<<edgeBlocks, 256, 0, stream>>>(sx, eB, srcI, dstI, ct, l, mx, num, den, E);
    // num <- aggr + sx   (in-place over num)
    k_aggr<<<2048, 256, 0, stream>>>(num, den, sx, num, N * 64);
    // inner MLP: relu(LN(num@W1+b1)) @ W2 + b2 ; l==0 overwrite h, else h += ...
    launch_gemm(stream, num, cW1 + (size_t)l * 64 * 128, cb1 + l * 128, t128,
                N, 64, 128, nullptr, nullptr, nullptr, 0.f, 0);
    k_ln_relu<<<N, 128, 0, stream>>>(t128, t128, cg + 128 * l, cB + 128 * l, 128);
    launch_gemm(stream, t128, cW2 + (size_t)l * 128 * 64, cb2 + l * 64, h,
                N, 128, 64, nullptr, nullptr, nullptr, 0.f, (l == 0) ? 0 : 4);
  }

  // ---- output MLP: 64 -> 32 (bn,relu) -> 16 (bn,relu) -> 2 ----
  launch_gemm(stream, h, oW1, ob1, t32, N, 64, 32, nullptr, nullptr, nullptr, 0.f, 0);
  hipMemsetAsync(stats0, 0, 2 * 32 * 4, stream);
  k_colstats<<<1024, 256, 0, stream>>>(t32, N, 32, stats0);
  launch_gemm(stream, t32, oW2, ob2, t16, N, 32, 16, stats0, og1, oB1, 1.f / (float)N, 1);
  hipMemsetAsync(stats1, 0, 2 * 16 * 4, stream);
  k_colstats<<<1024, 256, 0, stream>>>(t16, N, 16, stats1);
  k_final<<<(N + 255) / 256, 256, 0, stream>>>(t16, stats1, og2, oB2, oW3, ob3,
                                               (float*)d_out, N, 1.f / (float)N);
}